// Qeq_20581483282618
// MI455X (gfx1250) — compile-verified
//
#include <hip/hip_runtime.h>
#include <hip/hip_bf16.h>
#include <math.h>

typedef __attribute__((ext_vector_type(2))) float v2f;
typedef __attribute__((ext_vector_type(8))) float v8f;

#define N_ATOM 512
#define NB     32           // LU panel width (K per trailing update)
#define P      544          // 17 * 32, padded leading dimension
#define NWAVE  8            // 256 threads = 8 wave32
#define COULOMB_K 14.399645478425668f
#define SQRT_PI   1.7724538509055160f
#define SQRT2     1.4142135623730951f

// ---------------------------------------------------------------- chi = F @ w
__global__ void qeq_chi(const float* __restrict__ feats,
                        const float* __restrict__ w,
                        float* __restrict__ chi, int Ntot) {
    int i = blockIdx.x * blockDim.x + threadIdx.x;
    if (i >= Ntot) return;
    const float4* f4 = (const float4*)(feats + (size_t)i * 64);
    const float4* w4 = (const float4*)w;
    float s = 0.f;
#pragma unroll
    for (int j = 0; j < 16; ++j) {
        float4 a = f4[j], b = w4[j];
        s += a.x * b.x + a.y * b.y + a.z * b.z + a.w * b.w;
    }
    chi[i] = s;
}

// ------------------------------------------------- build padded bordered A
__global__ void qeq_build(const float* __restrict__ pos,
                          const int*   __restrict__ atype,
                          const float* __restrict__ hardness,
                          const float* __restrict__ sigma,
                          float* __restrict__ Aall) {
    const int r   = blockIdx.x;   // row 0..P-1
    const int mol = blockIdx.y;
    float* A = Aall + (size_t)mol * P * P;

    if (r < N_ATOM) {
        const int gi = mol * N_ATOM + r;
        const float px = pos[gi * 3 + 0];
        const float py = pos[gi * 3 + 1];
        const float pz = pos[gi * 3 + 2];
        const int   ti = atype[gi];
        const float si = sigma[ti];
        float hi = hardness[ti]; hi *= hi;
        for (int c = threadIdx.x; c < P; c += blockDim.x) {
            float val;
            if (c < N_ATOM) {
                if (c == r) {
                    val = hi + COULOMB_K / (SQRT_PI * si);
                } else {
                    const int gj = mol * N_ATOM + c;
                    float dx = px - pos[gj * 3 + 0];
                    float dy = py - pos[gj * 3 + 1];
                    float dz = pz - pos[gj * 3 + 2];
                    float d  = sqrtf(dx * dx + dy * dy + dz * dz);
                    float sj = sigma[atype[gj]];
                    float g  = sqrtf(si * si + sj * sj);
                    val = COULOMB_K * erff(d / (SQRT2 * g)) / d;
                }
            } else if (c == N_ATOM) {
                val = 1.0f;                       // border column
            } else {
                val = 0.0f;                       // padding
            }
            A[(size_t)r * P + c] = val;
        }
    } else if (r == N_ATOM) {                     // border row (charge constraint)
        for (int c = threadIdx.x; c < P; c += blockDim.x)
            A[(size_t)r * P + c] = (c < N_ATOM) ? 1.0f : 0.0f;
    } else {                                      // identity padding rows
        for (int c = threadIdx.x; c < P; c += blockDim.x)
            A[(size_t)r * P + c] = (c == r) ? 1.0f : 0.0f;
    }
}

// ------------------------------------------------------------------- rhs
__global__ void qeq_rhs(const float* __restrict__ chi,
                        const float* __restrict__ Qtot,
                        float* __restrict__ rhs) {
    const int mol = blockIdx.x;
    for (int i = threadIdx.x; i < P; i += blockDim.x) {
        float v = 0.f;
        if (i < N_ATOM)       v = -chi[mol * N_ATOM + i];
        else if (i == N_ATOM) v = Qtot[mol];
        rhs[mol * P + i] = v;
    }
}

// ------------------------------------- blocked LU (no pivot), WMMA trailing update
__global__ void __launch_bounds__(256) qeq_lu(float* __restrict__ Aall) {
    const int mol  = blockIdx.x;
    float* A = Aall + (size_t)mol * P * P;
    const int tid  = threadIdx.x;
    const int lane = tid & 31;
    const int wave = tid >> 5;

    __shared__ float sD[NB][NB];      // diagonal block (LU-factored)   4 KB
    __shared__ float sU[NB][P];       // U panel, absolute column idx  69.6 KB
    __shared__ float sLneg[P][NB];    // -L panel, absolute row idx    69.6 KB

    for (int k = 0; k < P; k += NB) {
        // ---- stage diagonal block (32x32)
        for (int idx = tid; idx < NB * NB; idx += 256)
            sD[idx >> 5][idx & 31] = A[(size_t)(k + (idx >> 5)) * P + k + (idx & 31)];
        __syncthreads();

        // ---- wave-parallel 32x32 Doolittle LU (unit lower), wave 0, lane = row.
        //      LDS ops within a wave are program-ordered, so the recurrence is safe.
        if (wave == 0) {
            for (int p = 0; p < NB; ++p) {
                float inv = 1.0f / sD[p][p];
                if (lane > p) {
                    float l = sD[lane][p] * inv;
                    sD[lane][p] = l;
                    for (int j = p + 1; j < NB; ++j)
                        sD[lane][j] -= l * sD[p][j];
                }
            }
        }
        __syncthreads();

        // ---- write factored diagonal back
        for (int idx = tid; idx < NB * NB; idx += 256)
            A[(size_t)(k + (idx >> 5)) * P + k + (idx & 31)] = sD[idx >> 5][idx & 31];

        // ---- U panel: forward-substitute each trailing column with unit-L
        for (int j = k + NB + tid; j < P; j += 256) {
            float x[NB];
#pragma unroll
            for (int i = 0; i < NB; ++i) x[i] = A[(size_t)(k + i) * P + j];
            for (int i = 1; i < NB; ++i) {
                float s = x[i];
                for (int t = 0; t < i; ++t) s -= sD[i][t] * x[t];
                x[i] = s;
            }
#pragma unroll
            for (int i = 0; i < NB; ++i) {
                A[(size_t)(k + i) * P + j] = x[i];
                sU[i][j] = x[i];
            }
        }

        // ---- L panel: right-solve each trailing row against U_d (store -L in LDS)
        for (int i = k + NB + tid; i < P; i += 256) {
            float x[NB];
#pragma unroll
            for (int t = 0; t < NB; ++t) x[t] = A[(size_t)i * P + k + t];
            for (int t = 0; t < NB; ++t) {
                float s = x[t];
                for (int u = 0; u < t; ++u) s -= x[u] * sD[u][t];
                x[t] = s / sD[t][t];
            }
#pragma unroll
            for (int t = 0; t < NB; ++t) {
                A[(size_t)i * P + k + t] = x[t];
                sLneg[i][t] = -x[t];
            }
        }
        __syncthreads();

        // ---- trailing update: C += (-L) * U, 8 chained V_WMMA_F32_16X16X4_F32
        //      per 16x16 C tile (K = 32) -> 8 WMMAs per 8 loads + 8 stores.
        const int base = k + NB;
        const int nt   = (P - base) >> 4;
        const int m  = lane & 15;      // column within tile (B/C/D layout)
        const int hs = lane >> 4;      // half-select
        for (int t = wave; t < nt * nt; t += NWAVE) {
            const int ti = t / nt, tj = t % nt;
            const int r0 = base + ti * 16;
            const int c0 = base + tj * 16;

            // prefetch the C tile this wave handles next
            int tn = t + NWAVE;
            if (tn < nt * nt) {
                int rn = base + (tn / nt) * 16, cn = base + (tn % nt) * 16;
                __builtin_prefetch(&A[(size_t)rn * P + cn], 0, 3);
            }

            v8f c;
#pragma unroll
            for (int v = 0; v < 8; ++v)
                c[v] = A[(size_t)(r0 + v + 8 * hs) * P + c0 + m];

#pragma unroll
            for (int cc = 0; cc < NB / 4; ++cc) {
                const int kb = 4 * cc + 2 * hs;
                v2f a, b;
                a[0] = sLneg[r0 + m][kb + 0];
                a[1] = sLneg[r0 + m][kb + 1];
                b[0] = sU[kb + 0][c0 + m];
                b[1] = sU[kb + 1][c0 + m];
                // f32 WMMA: A/B negation illegal (ISA 7.12), so A holds -L.
                c = __builtin_amdgcn_wmma_f32_16x16x4_f32(
                        false, a, false, b, (short)0, c, false, false);
            }
#pragma unroll
            for (int v = 0; v < 8; ++v)
                A[(size_t)(r0 + v + 8 * hs) * P + c0 + m] = c[v];
        }
        __threadfence();
        __syncthreads();
    }
}

// --------------------------------------------- triangular solves, one WG / molecule
__global__ void __launch_bounds__(256) qeq_solve(const float* __restrict__ Aall,
                                                 const float* __restrict__ rhs,
                                                 float* __restrict__ qout,
                                                 float* __restrict__ lam) {
    const int mol = blockIdx.x;
    const float* A = Aall + (size_t)mol * P * P;
    __shared__ float y[P];
    for (int i = threadIdx.x; i < P; i += blockDim.x) y[i] = rhs[mol * P + i];
    __syncthreads();
    // forward substitution, unit-lower L
    for (int j = 0; j < P - 1; ++j) {
        const float yj = y[j];
        for (int i = j + 1 + threadIdx.x; i < P; i += blockDim.x)
            y[i] -= A[(size_t)i * P + j] * yj;
        __syncthreads();
    }
    // back substitution, upper U
    for (int j = P - 1; j >= 0; --j) {
        if (threadIdx.x == 0) y[j] /= A[(size_t)j * P + j];
        __syncthreads();
        const float xj = y[j];
        for (int i = threadIdx.x; i < j; i += blockDim.x)
            y[i] -= A[(size_t)i * P + j] * xj;
        __syncthreads();
    }
    for (int i = threadIdx.x; i < N_ATOM; i += blockDim.x)
        qout[mol * N_ATOM + i] = y[i];
    if (threadIdx.x == 0) lam[mol] = y[N_ATOM];
}

// -------------------- energy: e = 0.5*chi.q - 0.5*lambda*Q  (bordered-row identity)
__global__ void __launch_bounds__(256) qeq_energy(const float* __restrict__ chi,
                                                  const float* __restrict__ q,
                                                  const float* __restrict__ lam,
                                                  const float* __restrict__ Qtot,
                                                  float* __restrict__ e) {
    const int mol = blockIdx.x;
    __shared__ float red[256];
    float s = 0.f;
    for (int i = threadIdx.x; i < N_ATOM; i += blockDim.x)
        s += chi[mol * N_ATOM + i] * q[mol * N_ATOM + i];
    red[threadIdx.x] = s;
    __syncthreads();
    for (int st = 128; st > 0; st >>= 1) {
        if (threadIdx.x < st) red[threadIdx.x] += red[threadIdx.x + st];
        __syncthreads();
    }
    if (threadIdx.x == 0)
        e[mol] = 0.5f * red[0] - 0.5f * lam[mol] * Qtot[mol];
}

// ------------------------------------------------------------------ launcher
extern "C" void kernel_launch(void* const* d_in, const int* in_sizes, int n_in,
                              void* d_out, int out_size, void* d_ws, size_t ws_size,
                              hipStream_t stream) {
    const float* feats = (const float*)d_in[0];   // (N, 64)
    const float* pos   = (const float*)d_in[1];   // (N, 3)
    const int*   atype = (const int*)  d_in[2];   // (N,)
    const float* Qtot  = (const float*)d_in[3];   // (B, 1)
    const float* w     = (const float*)d_in[4];   // (64, 1)
    const float* hard  = (const float*)d_in[5];   // (S,)
    const float* sigma = (const float*)d_in[6];   // (S,)

    const int B = in_sizes[3];                    // 128 molecules
    const int Ntot = B * N_ATOM;                  // 65536 atoms

    // workspace layout (floats)
    float* A   = (float*)d_ws;                    // B * P * P
    float* chi = A   + (size_t)B * P * P;         // Ntot
    float* rhs = chi + (size_t)Ntot;              // B * P
    float* lam = rhs + (size_t)B * P;             // B

    float* e_out = (float*)d_out;                 // (B,)
    float* q_out = e_out + B;                     // (Ntot,)

    qeq_chi  <<<(Ntot + 255) / 256, 256, 0, stream>>>(feats, w, chi, Ntot);
    qeq_build<<<dim3(P, B), 256, 0, stream>>>(pos, atype, hard, sigma, A);
    qeq_rhs  <<<B, 256, 0, stream>>>(chi, Qtot, rhs);
    qeq_lu   <<<B, 256, 0, stream>>>(A);
    qeq_solve<<<B, 256, 0, stream>>>(A, rhs, q_out, lam);
    qeq_energy<<<B, 256, 0, stream>>>(chi, q_out, lam, Qtot, e_out);
}